// ConditionalRandomField_35029753266239
// MI455X (gfx1250) — compile-verified
//
#include <hip/hip_runtime.h>
#include <hip/hip_bf16.h>

typedef __attribute__((ext_vector_type(2))) float v2f;
typedef __attribute__((ext_vector_type(8))) float v8f;

#define B_   256
#define L_   256
#define T_   50
#define NP   64     // padded N (tags, columns of E)
#define KP   52     // padded K (rows of E), multiple of 4
#define MT   16     // batch rows per workgroup (WMMA M)
#define NTHR 128    // 4 waves: one 16-wide N tile each

// Forward algorithm: one WG handles 16 batch rows across all 256 time steps.
// Step recurrence done as  alpha' = log( exp(alpha - rowmax) * exp(trans) ) + rowmax + emit
// with the GEMM on v_wmma_f32_16x16x4_f32 and exp(trans) fragments resident in VGPRs.
__global__ __launch_bounds__(NTHR) void crf_forward_kernel(
    const float* __restrict__ feats, const float* __restrict__ trans,
    const int* __restrict__ mask, float* __restrict__ allpath)
{
  __shared__ float Esh[KP][NP];     // exp(trans), zero padded
  __shared__ float alpha[MT][NP];   // log-domain state, -1e30 in padding
  __shared__ float vbuf[MT][NP];    // exp(alpha - rowmax)
  __shared__ float pmax[MT][8];
  __shared__ float rowmax[MT];

  const int tid  = threadIdx.x;
  const int lane = tid & 31;
  const int wv   = tid >> 5;               // wave id == N tile index
  const int b0   = blockIdx.x * MT;

  // Build E = exp(trans) padded to [52][64]
  for (int idx = tid; idx < KP * NP; idx += NTHR) {
    int k = idx >> 6, n = idx & 63;
    float e = 0.0f;
    if (k < T_ && n < T_) e = __expf(trans[k * T_ + n]);
    Esh[k][n] = e;
  }
  // alpha0 = feats[:, 0, :]
  for (int idx = tid; idx < MT * NP; idx += NTHR) {
    int m = idx >> 6, n = idx & 63;
    alpha[m][n] = (n < T_) ? feats[((b0 + m) * L_) * T_ + n] : -1e30f;
  }
  __syncthreads();

  // Preload loop-invariant B fragments (exp(trans)) into registers.
  // B 4x16 layout: VGPR0 = rows K0 (lanes0-15) / K2 (lanes16-31), VGPR1 = K1 / K3.
  const int khalf = (lane >> 4) * 2;
  const int nloc  = wv * 16 + (lane & 15); // this lane's output column
  v2f bfrag[13];
  #pragma unroll
  for (int kb = 0; kb < 13; ++kb) {
    bfrag[kb].x = Esh[kb * 4 + khalf][nloc];
    bfrag[kb].y = Esh[kb * 4 + khalf + 1][nloc];
  }

  const int row   = tid >> 3;       // reduction helper indices
  const int c0    = (tid & 7) * 8;
  const int arow  = lane & 15;      // A-matrix row for this lane
  const int mbase = 8 * (lane >> 4);// D rows owned: mbase..mbase+7

  for (int t = 1; t < L_; ++t) {
    // Prefetch emissions + mask for the D elements this lane owns (overlaps reduction)
    float emitr[8];
    int   maskr[8];
    if (nloc < T_) {
      #pragma unroll
      for (int r = 0; r < 8; ++r) {
        int b = b0 + mbase + r;
        emitr[r] = feats[(b * L_ + t) * T_ + nloc];
        maskr[r] = mask[b * L_ + t];
      }
    }
    // Row-max reduction (stability): 8 partials per row, then 16 threads finish
    float mx = alpha[row][c0];
    #pragma unroll
    for (int c = 1; c < 8; ++c) mx = fmaxf(mx, alpha[row][c0 + c]);
    pmax[row][tid & 7] = mx;
    __syncthreads();
    if (tid < MT) {
      float m2 = pmax[tid][0];
      #pragma unroll
      for (int c = 1; c < 8; ++c) m2 = fmaxf(m2, pmax[tid][c]);
      rowmax[tid] = m2;
    }
    __syncthreads();
    // v = exp(alpha - rowmax)  (padding -> exp(-huge) = 0)
    {
      float rm = rowmax[row];
      #pragma unroll
      for (int c = 0; c < 8; ++c)
        vbuf[row][c0 + c] = __expf(alpha[row][c0 + c] - rm);
    }
    __syncthreads();
    // S = v * E : 13 chained f32 WMMAs (K blocks of 4)
    // A 16x4 layout: lanes0-15 hold {K0,K1}, lanes16-31 hold {K2,K3} per block.
    v8f acc = {};
    #pragma unroll
    for (int kb = 0; kb < 13; ++kb) {
      v2f a = *(const v2f*)&vbuf[arow][kb * 4 + khalf];
      acc = __builtin_amdgcn_wmma_f32_16x16x4_f32(
          false, a, false, bfrag[kb], (short)0, acc, false, false);
    }
    // alpha' = log(S) + rowmax + emit  (masked update); lane owns (M, nloc) exclusively
    if (nloc < T_) {
      #pragma unroll
      for (int r = 0; r < 8; ++r) {
        int M = mbase + r;
        float oldv = alpha[M][nloc];
        float nv   = __logf(acc[r]) + rowmax[M] + emitr[r];
        alpha[M][nloc] = maskr[r] ? nv : oldv;
      }
    }
    __syncthreads();
  }

  // Final per-row logsumexp -> allpath
  {
    float mx = alpha[row][c0];
    #pragma unroll
    for (int c = 1; c < 8; ++c) mx = fmaxf(mx, alpha[row][c0 + c]);
    pmax[row][tid & 7] = mx;
  }
  __syncthreads();
  if (tid < MT) {
    float m2 = pmax[tid][0];
    #pragma unroll
    for (int c = 1; c < 8; ++c) m2 = fmaxf(m2, pmax[tid][c]);
    float s = 0.0f;
    for (int n = 0; n < T_; ++n) s += __expf(alpha[tid][n] - m2);
    allpath[b0 + tid] = m2 + __logf(s);
  }
}

// Gold path score + final subtraction: one thread per batch element.
__global__ __launch_bounds__(256) void crf_gold_kernel(
    const float* __restrict__ feats, const float* __restrict__ trans,
    const int* __restrict__ tags, const int* __restrict__ mask,
    const float* __restrict__ allpath, float* __restrict__ out)
{
  int b = blockIdx.x * blockDim.x + threadIdx.x;
  if (b >= B_) return;
  int prev = tags[b * L_];
  float gold = feats[(b * L_) * T_ + prev] * (float)mask[b * L_];
  for (int t = 1; t < L_; ++t) {
    int cur  = tags[b * L_ + t];
    float mt = (float)mask[b * L_ + t];
    gold += (trans[prev * T_ + cur] + feats[(b * L_ + t) * T_ + cur]) * mt;
    prev = cur;
  }
  out[b] = allpath[b] - gold;
}

extern "C" void kernel_launch(void* const* d_in, const int* in_sizes, int n_in,
                              void* d_out, int out_size, void* d_ws, size_t ws_size,
                              hipStream_t stream) {
  const float* feats = (const float*)d_in[0];
  const float* trans = (const float*)d_in[1];
  const int*   tags  = (const int*)d_in[2];
  const int*   mask  = (const int*)d_in[3];
  float* out     = (float*)d_out;
  float* allpath = (float*)d_ws;   // 256 floats of scratch

  crf_forward_kernel<<<B_ / MT, NTHR, 0, stream>>>(feats, trans, mask, allpath);
  crf_gold_kernel<<<1, 256, 0, stream>>>(feats, trans, tags, mask, allpath, out);
}